// Memory_21071109554526
// MI455X (gfx1250) — compile-verified
//
#include <hip/hip_runtime.h>
#include <hip/hip_bf16.h>

typedef __attribute__((ext_vector_type(2))) float v2f;
typedef __attribute__((ext_vector_type(8))) float v8f;

#define HID   256
#define BS    64
#define DCS   16
#define NCLS  60
#define QS    5
#define NROW  (BS * DCS)        // 1024 rows of h
#define MKROW (NCLS * DCS)      // 960 memory-key rows
#define KQROW (NCLS * QS * DCS) // 4800 queue-key rows
#define NCAND (QS * QS * DCS)   // 400 candidates per row
#define TOPKQ 100

__device__ __forceinline__ v8f wmma4(v2f a, v2f b, v8f c) {
  // D = A(16x4,f32) * B(4x16,f32) + C(16x16,f32)
  return __builtin_amdgcn_wmma_f32_16x16x4_f32(false, a, false, b, (short)0, c,
                                               false, false);
}

// ---------------------------------------------------------------------------
// Y[M,N] = X[M,K=256] @ W[N,K=256]^T (+ bias[N]) ; row-major.
// One wave computes a 16(M) x 64(N) tile with 4 accumulators.
// K is constexpr so all fragment loads use immediate offsets off two base
// pointers; explicit software pipeline (load k+4 before WMMAs of k) lets the
// scheduler overlap loads with WMMA instead of s_wait_loadcnt 0x0 per op.
// Requires (M/16)*(N/64) waves, launched exactly (8 waves / 256-thr block).
// ---------------------------------------------------------------------------
__global__ __launch_bounds__(256) void gemm_xwt_wmma(
    const float* __restrict__ X, const float* __restrict__ W,
    const float* __restrict__ bias, float* __restrict__ Y,
    int M, int N) {
  constexpr int K = HID;
  int wave  = (int)((blockIdx.x * blockDim.x + threadIdx.x) >> 5);
  int strips = N >> 6;                 // N / 64
  int tm  = wave / strips;
  int tn4 = wave - tm * strips;        // 64-wide N strip
  int lane = threadIdx.x & 31;
  int half = lane >> 4;                // 0: K pair {0,1}, 1: K pair {2,3}
  int l16  = lane & 15;

  const float* xr = X + (size_t)(tm * 16 + l16) * K + half * 2;
  const float* wr = W + (size_t)(tn4 * 64 + l16) * K + half * 2;

  v8f acc0 = {}, acc1 = {}, acc2 = {}, acc3 = {};

  v2f a  = *(const v2f*)(xr);
  v2f b0 = *(const v2f*)(wr);
  v2f b1 = *(const v2f*)(wr + 16 * K);
  v2f b2 = *(const v2f*)(wr + 32 * K);
  v2f b3 = *(const v2f*)(wr + 48 * K);
#pragma unroll 3
  for (int k = 0; k < K - 4; k += 4) {
    v2f an  = *(const v2f*)(xr + k + 4);
    v2f bn0 = *(const v2f*)(wr + k + 4);
    v2f bn1 = *(const v2f*)(wr + k + 4 + 16 * K);
    v2f bn2 = *(const v2f*)(wr + k + 4 + 32 * K);
    v2f bn3 = *(const v2f*)(wr + k + 4 + 48 * K);
    acc0 = wmma4(a, b0, acc0);
    acc1 = wmma4(a, b1, acc1);
    acc2 = wmma4(a, b2, acc2);
    acc3 = wmma4(a, b3, acc3);
    a = an; b0 = bn0; b1 = bn1; b2 = bn2; b3 = bn3;
  }
  acc0 = wmma4(a, b0, acc0);
  acc1 = wmma4(a, b1, acc1);
  acc2 = wmma4(a, b2, acc2);
  acc3 = wmma4(a, b3, acc3);

  float* ybase = Y + (size_t)(tm * 16 + half * 8) * N + tn4 * 64 + l16;
  v8f accs[4] = {acc0, acc1, acc2, acc3};
#pragma unroll
  for (int s = 0; s < 4; s++) {
    float bb = bias ? bias[tn4 * 64 + s * 16 + l16] : 0.f;
    float* yp = ybase + s * 16;
#pragma unroll
    for (int v = 0; v < 8; v++) yp[(size_t)v * N] = accs[s][v] + bb;
  }
}

// ---------------------------------------------------------------------------
// Final fused GEMM: z = concat(queue, memory_z) @ proto_w^T + proto_b
// M=1024, N=256, K=512 (first 256 from queue, next 256 from memory_z).
// Same 16x64 blocking + software pipeline; ldw = 512 constexpr.
// ---------------------------------------------------------------------------
__global__ __launch_bounds__(256) void concat_gemm_wmma(
    const float* __restrict__ Aq, const float* __restrict__ Am,
    const float* __restrict__ W, const float* __restrict__ bias,
    float* __restrict__ Y) {
  constexpr int LDW = 2 * HID;
  int wave = (int)((blockIdx.x * blockDim.x + threadIdx.x) >> 5);
  int tm  = wave >> 2;                 // N/64 == 4 strips
  int tn4 = wave & 3;
  int lane = threadIdx.x & 31;
  int half = lane >> 4;
  int l16  = lane & 15;

  const float* q  = Aq + (size_t)(tm * 16 + l16) * HID + half * 2;
  const float* mz = Am + (size_t)(tm * 16 + l16) * HID + half * 2;
  const float* wr = W + (size_t)(tn4 * 64 + l16) * LDW + half * 2;

  v8f acc0 = {}, acc1 = {}, acc2 = {}, acc3 = {};

  // ---- half 1: queue columns (W cols 0..255) ----
  v2f a  = *(const v2f*)(q);
  v2f b0 = *(const v2f*)(wr);
  v2f b1 = *(const v2f*)(wr + 16 * LDW);
  v2f b2 = *(const v2f*)(wr + 32 * LDW);
  v2f b3 = *(const v2f*)(wr + 48 * LDW);
#pragma unroll 3
  for (int k = 0; k < HID - 4; k += 4) {
    v2f an  = *(const v2f*)(q + k + 4);
    v2f bn0 = *(const v2f*)(wr + k + 4);
    v2f bn1 = *(const v2f*)(wr + k + 4 + 16 * LDW);
    v2f bn2 = *(const v2f*)(wr + k + 4 + 32 * LDW);
    v2f bn3 = *(const v2f*)(wr + k + 4 + 48 * LDW);
    acc0 = wmma4(a, b0, acc0);
    acc1 = wmma4(a, b1, acc1);
    acc2 = wmma4(a, b2, acc2);
    acc3 = wmma4(a, b3, acc3);
    a = an; b0 = bn0; b1 = bn1; b2 = bn2; b3 = bn3;
  }
  acc0 = wmma4(a, b0, acc0);
  acc1 = wmma4(a, b1, acc1);
  acc2 = wmma4(a, b2, acc2);
  acc3 = wmma4(a, b3, acc3);

  // ---- half 2: memory_z columns (W cols 256..511) ----
  a  = *(const v2f*)(mz);
  b0 = *(const v2f*)(wr + HID);
  b1 = *(const v2f*)(wr + HID + 16 * LDW);
  b2 = *(const v2f*)(wr + HID + 32 * LDW);
  b3 = *(const v2f*)(wr + HID + 48 * LDW);
#pragma unroll 3
  for (int k = 0; k < HID - 4; k += 4) {
    v2f an  = *(const v2f*)(mz + k + 4);
    v2f bn0 = *(const v2f*)(wr + HID + k + 4);
    v2f bn1 = *(const v2f*)(wr + HID + k + 4 + 16 * LDW);
    v2f bn2 = *(const v2f*)(wr + HID + k + 4 + 32 * LDW);
    v2f bn3 = *(const v2f*)(wr + HID + k + 4 + 48 * LDW);
    acc0 = wmma4(a, b0, acc0);
    acc1 = wmma4(a, b1, acc1);
    acc2 = wmma4(a, b2, acc2);
    acc3 = wmma4(a, b3, acc3);
    a = an; b0 = bn0; b1 = bn1; b2 = bn2; b3 = bn3;
  }
  acc0 = wmma4(a, b0, acc0);
  acc1 = wmma4(a, b1, acc1);
  acc2 = wmma4(a, b2, acc2);
  acc3 = wmma4(a, b3, acc3);

  float* ybase = Y + (size_t)(tm * 16 + half * 8) * HID + tn4 * 64 + l16;
  v8f accs[4] = {acc0, acc1, acc2, acc3};
#pragma unroll
  for (int s = 0; s < 4; s++) {
    float bb = bias[tn4 * 64 + s * 16 + l16];
    float* yp = ybase + s * 16;
#pragma unroll
    for (int v = 0; v < 8; v++) yp[(size_t)v * HID] = accs[s][v] + bb;
  }
}

// ---------------------------------------------------------------------------
// Top-5 over 960 per row + softmax. Stable (first-occurrence) tie-break.
// ---------------------------------------------------------------------------
__global__ __launch_bounds__(256) void topk5_softmax(
    const float* __restrict__ Wt, float* __restrict__ top_w,
    int* __restrict__ top_i) {
  int n = blockIdx.x * blockDim.x + threadIdx.x;
  if (n >= NROW) return;
  const float* row = Wt + (size_t)n * MKROW;
  float v[5]; int id[5];
#pragma unroll
  for (int j = 0; j < 5; j++) { v[j] = -3.402823466e38f; id[j] = 0; }
  for (int j = 0; j < MKROW; j++) {
    float x = row[j];
    if (x > v[4]) {
      int p = 4;
      while (p > 0 && x > v[p - 1]) { v[p] = v[p - 1]; id[p] = id[p - 1]; p--; }
      v[p] = x; id[p] = j;
    }
  }
  float m = v[0], s = 0.f, e[5];
#pragma unroll
  for (int j = 0; j < 5; j++) { e[j] = expf(v[j] - m); s += e[j]; }
  float inv = 1.f / s;
#pragma unroll
  for (int j = 0; j < 5; j++) { top_w[n * 5 + j] = e[j] * inv; top_i[n * 5 + j] = id[j]; }
}

// memory_z[n][c] = sum_j top_w[n][j] * key[top_i[n][j]][c]
__global__ __launch_bounds__(256) void gather_memz(
    const float* __restrict__ key, const float* __restrict__ top_w,
    const int* __restrict__ top_i, float* __restrict__ memz) {
  int n = blockIdx.x, c = threadIdx.x;
  float acc = 0.f;
#pragma unroll
  for (int j = 0; j < 5; j++)
    acc += top_w[n * 5 + j] * key[(size_t)top_i[n * 5 + j] * HID + c];
  memz[(size_t)n * HID + c] = acc;
}

// ---------------------------------------------------------------------------
// w_queue[n][m] = query_q[n] . kq_all[class(n, m/80)*80 + m%80]
// one block per row; query row staged in LDS; wave-per-dot shuffle reduction
// ---------------------------------------------------------------------------
__global__ __launch_bounds__(256) void wqueue_kernel(
    const float* __restrict__ qq, const float* __restrict__ kqa,
    const int* __restrict__ top_i, float* __restrict__ wq) {
  __shared__ float q[HID];
  int n = blockIdx.x;
  q[threadIdx.x] = qq[(size_t)n * HID + threadIdx.x];
  __syncthreads();
  int wave = threadIdx.x >> 5, lane = threadIdx.x & 31;
  int cls[5];
#pragma unroll
  for (int j = 0; j < 5; j++) cls[j] = top_i[n * 5 + j] >> 4;  // // DCS
  for (int m = wave; m < NCAND; m += 8) {
    int j = m / 80, r = m - j * 80;
    const float* krow = kqa + (size_t)(cls[j] * 80 + r) * HID;
    float s = 0.f;
#pragma unroll
    for (int i = 0; i < HID / 32; i++) s += krow[lane + 32 * i] * q[lane + 32 * i];
    for (int off = 16; off; off >>= 1) s += __shfl_xor(s, off, 32);
    if (lane == 0) wq[(size_t)n * NCAND + m] = s;
  }
}

// ---------------------------------------------------------------------------
// Top-100 of 400 via rank counting (exact, stable), softmax, weighted sum.
// ---------------------------------------------------------------------------
__global__ __launch_bounds__(256) void topq_queue(
    const float* __restrict__ wq, const float* __restrict__ kqa,
    const int* __restrict__ top_i, float* __restrict__ queue) {
  __shared__ float vals[NCAND];
  __shared__ float wts[NCAND];
  __shared__ int   sel[TOPKQ];
  __shared__ int   cnt;
  __shared__ float ssum;
  __shared__ float smax;
  __shared__ float wmax[8];
  int n = blockIdx.x, t = threadIdx.x;
  if (t == 0) { cnt = 0; ssum = 0.f; }
  for (int m = t; m < NCAND; m += 256) vals[m] = wq[(size_t)n * NCAND + m];
  __syncthreads();
  // row max (always in the selected set)
  float lm = -3.402823466e38f;
  for (int m = t; m < NCAND; m += 256) lm = fmaxf(lm, vals[m]);
  for (int off = 16; off; off >>= 1) lm = fmaxf(lm, __shfl_xor(lm, off, 32));
  if ((t & 31) == 0) wmax[t >> 5] = lm;
  __syncthreads();
  if (t == 0) {
    float mm = wmax[0];
#pragma unroll
    for (int i = 1; i < 8; i++) mm = fmaxf(mm, wmax[i]);
    smax = mm;
  }
  __syncthreads();
  // strict total-order rank => exactly TOPKQ selected, matches stable top_k
  for (int m = t; m < NCAND; m += 256) {
    float x = vals[m];
    int rank = 0;
    for (int j = 0; j < NCAND; j++)
      rank += (vals[j] > x) || (vals[j] == x && j < m);
    float e = 0.f;
    if (rank < TOPKQ) {
      e = expf(x - smax);
      int p = atomicAdd(&cnt, 1);
      sel[p] = m;
      atomicAdd(&ssum, e);
    }
    wts[m] = e;
  }
  __syncthreads();
  float inv = 1.f / ssum;
  int cls[5];
#pragma unroll
  for (int j = 0; j < 5; j++) cls[j] = top_i[n * 5 + j] >> 4;
  float acc = 0.f;
  for (int s = 0; s < TOPKQ; s++) {
    int m = sel[s];
    int j = m / 80, r = m - j * 80;
    acc += wts[m] * kqa[(size_t)(cls[j] * 80 + r) * HID + t];
  }
  queue[(size_t)n * HID + t] = acc * inv;
}

// ---------------------------------------------------------------------------
extern "C" void kernel_launch(void* const* d_in, const int* in_sizes, int n_in,
                              void* d_out, int out_size, void* d_ws, size_t ws_size,
                              hipStream_t stream) {
  const float* h          = (const float*)d_in[0];
  // d_in[1] labels, d_in[2] tag: unused by the reference math
  const float* memory_key = (const float*)d_in[3];
  const float* queue_key  = (const float*)d_in[4];
  const float* key_w   = (const float*)d_in[5];
  const float* key_b   = (const float*)d_in[6];
  const float* query_w = (const float*)d_in[7];
  const float* query_b = (const float*)d_in[8];
  const float* kq_w    = (const float*)d_in[9];
  const float* kq_b    = (const float*)d_in[10];
  const float* qq_w    = (const float*)d_in[11];
  const float* qq_b    = (const float*)d_in[12];
  const float* proto_w = (const float*)d_in[13];
  const float* proto_b = (const float*)d_in[14];
  float* out = (float*)d_out;

  // workspace carve-up (floats), ~15.7 MB total
  float* ws = (float*)d_ws;
  float* key_buf  = ws;                              // 960*256
  float* query    = key_buf  + MKROW * HID;          // 1024*256
  float* query_q  = query    + NROW * HID;           // 1024*256
  float* kq_all   = query_q  + NROW * HID;           // 4800*256
  float* weights  = kq_all   + KQROW * HID;          // 1024*960
  float* memz     = weights  + (size_t)NROW * MKROW; // 1024*256
  float* wqueue   = memz     + NROW * HID;           // 1024*400
  float* queue    = wqueue   + (size_t)NROW * NCAND; // 1024*256
  float* top_w    = queue    + NROW * HID;           // 1024*5
  int*   top_i    = (int*)(top_w + NROW * 5);        // 1024*5

  // 16x64 tile per wave, 8 waves per 256-thread block
  auto blocks = [](int M, int N) { return ((M >> 4) * (N >> 6)) >> 3; };

  // level-1 linears (WMMA f32)
  gemm_xwt_wmma<<<blocks(MKROW, HID), 256, 0, stream>>>(memory_key, key_w, key_b, key_buf, MKROW, HID);
  gemm_xwt_wmma<<<blocks(NROW, HID), 256, 0, stream>>>(h, query_w, query_b, query, NROW, HID);
  gemm_xwt_wmma<<<blocks(NROW, HID), 256, 0, stream>>>(h, qq_w, qq_b, query_q, NROW, HID);
  // hoisted queue-key linear over the 4800 distinct rows
  gemm_xwt_wmma<<<blocks(KQROW, HID), 256, 0, stream>>>(queue_key, kq_w, kq_b, kq_all, KQROW, HID);
  // weights = query @ key^T  (N=960 -> 15 strips of 64)
  gemm_xwt_wmma<<<blocks(NROW, MKROW), 256, 0, stream>>>(query, key_buf, nullptr, weights, NROW, MKROW);

  topk5_softmax<<<NROW / 256, 256, 0, stream>>>(weights, top_w, top_i);
  gather_memz<<<NROW, 256, 0, stream>>>(key_buf, top_w, top_i, memz);
  wqueue_kernel<<<NROW, 256, 0, stream>>>(query_q, kq_all, top_i, wqueue);
  topq_queue<<<NROW, 256, 0, stream>>>(wqueue, kq_all, top_i, queue);

  // z = concat(queue, memz) @ proto_w^T + proto_b
  concat_gemm_wmma<<<blocks(NROW, HID), 256, 0, stream>>>(queue, memz, proto_w, proto_b, out);
}